// ModelExperts_5669356832630
// MI455X (gfx1250) — compile-verified
//
#include <hip/hip_runtime.h>

// MoE grouped GEMM for gfx1250 (MI455X), f32 via V_WMMA_F32_16X16X4_F32.
// out[t] = x[t] @ W[index[t]] + b[index[t]]
//
// Pipeline: histogram -> 16-aligned prefix -> scatter token ids ->
// grouped GEMM over 16-token tiles (one expert per tile, padded with -1).
// Each wave computes a 16x32 output tile as TWO independent WMMA
// accumulator chains, so the XDL pipe always has an independent op to issue.

#define N_TOKENS   32768
#define N_EXPERTS  8
#define K_DIM      512
#define N_DIM      512
#define TILE_M     16
#define PERM_CAP   (N_TOKENS + N_EXPERTS * TILE_M)     // 32896 (worst-case padding)
#define MAX_TILES  (N_TOKENS / TILE_M + N_EXPERTS)     // 2056
#define LDSX       516   // row stride in floats: 16B aligned rows, conflict-free b64 reads

typedef __attribute__((ext_vector_type(2))) float v2f;
typedef __attribute__((ext_vector_type(8))) float v8f;

// ---- workspace int layout -------------------------------------------------
//  ws[0..8]    : off[9]     (16-aligned exclusive prefix, off[8] = padded total)
//  ws[16..23]  : cursor[8]  (scatter cursors)
//  ws[32..39]  : counts[8]
//  ws[64..]    : perm[PERM_CAP]  (token ids grouped by expert, -1 = pad)

__global__ void moe_init(int* __restrict__ counts, int* __restrict__ perm) {
    const int i = blockIdx.x * blockDim.x + threadIdx.x;
    if (i < PERM_CAP) perm[i] = -1;
    if (i < N_EXPERTS) counts[i] = 0;
}

__global__ void moe_count(const int* __restrict__ index, int* __restrict__ counts) {
    const int t = blockIdx.x * blockDim.x + threadIdx.x;
    if (t < N_TOKENS) atomicAdd(&counts[index[t] & (N_EXPERTS - 1)], 1);
}

__global__ void moe_prefix(int* __restrict__ ws) {
    if (threadIdx.x == 0 && blockIdx.x == 0) {
        int* off          = ws;
        int* cursor       = ws + 16;
        const int* counts = ws + 32;
        int acc = 0;
        for (int e = 0; e < N_EXPERTS; ++e) {
            off[e]    = acc;
            cursor[e] = acc;
            acc += (counts[e] + TILE_M - 1) & ~(TILE_M - 1);
        }
        off[N_EXPERTS] = acc;
    }
}

__global__ void moe_scatter(const int* __restrict__ index, int* __restrict__ cursor,
                            int* __restrict__ perm) {
    const int t = blockIdx.x * blockDim.x + threadIdx.x;
    if (t < N_TOKENS) {
        const int e = index[t] & (N_EXPERTS - 1);
        const int p = atomicAdd(&cursor[e], 1);
        perm[p] = t;
    }
}

// One block = one 16-token tile x 256 output columns.  8 waves; each wave
// owns a 32-column strip: two independent 16x16 accumulator chains,
// 2 x 128 WMMAs over K=512, sharing one A fragment per step.
__global__ __launch_bounds__(256) void moe_gemm(
    const float* __restrict__ x, const float* __restrict__ weight,
    const float* __restrict__ bias, const int* __restrict__ off,
    const int* __restrict__ perm, float* __restrict__ out)
{
    __shared__ int   s_off[N_EXPERTS + 1];
    __shared__ int   s_tok[TILE_M];
    __shared__ float s_x[TILE_M * LDSX];

    if (threadIdx.x <= N_EXPERTS) s_off[threadIdx.x] = off[threadIdx.x];
    __syncthreads();

    const int base = blockIdx.x * TILE_M;
    if (base >= s_off[N_EXPERTS]) return;            // block-uniform exit

    int e = 0;                                       // segments are 16-aligned,
    while (base >= s_off[e + 1]) ++e;                // so a tile has one expert

    if (threadIdx.x < TILE_M) s_tok[threadIdx.x] = perm[base + threadIdx.x];
    __syncthreads();

    // Gather 16 token rows of x into LDS (pad rows -> zeros).
    for (int idx = threadIdx.x; idx < TILE_M * (K_DIM / 4); idx += blockDim.x) {
        const int r   = idx >> 7;                    // 0..15
        const int c4  = (idx & 127) << 2;            // 0..508 step 4
        const int tok = s_tok[r];
        float4 v = make_float4(0.f, 0.f, 0.f, 0.f);
        if (tok >= 0) v = *(const float4*)(x + (size_t)tok * K_DIM + c4);
        *(float4*)(&s_x[r * LDSX + c4]) = v;
    }
    __syncthreads();

    const int lane = threadIdx.x & 31;
    const int wv   = threadIdx.x >> 5;
    const int mrow = lane & 15;                      // A: M = lane%16
    const int koff = (lane >> 4) << 1;               // A/B: K base 0 or 2 per half-wave
    const int ncol = lane & 15;                      // B/C/D: N = lane%16
    const int n0   = blockIdx.y * 256 + wv * 32;     // this wave's 32-col strip

    // W[k][n] = Wp[k*N_DIM]; two column strips 16 apart
    const float* Wp0 = weight + (size_t)e * K_DIM * N_DIM + n0 + ncol;
    const float* Wp1 = Wp0 + 16;
    const float* Ap  = &s_x[mrow * LDSX];

    v8f acc0 = {};
    v8f acc1 = {};
    for (int kb = 0; kb < K_DIM; kb += 16) {
        if (kb + 16 < K_DIM) {                       // uniform branch (EXEC intact)
            __builtin_prefetch(Wp0 + (size_t)(kb + 16 + koff) * N_DIM, 0, 3);
            __builtin_prefetch(Wp1 + (size_t)(kb + 16 + koff) * N_DIM, 0, 3);
        }
#pragma unroll
        for (int s = 0; s < 4; ++s) {
            const int k = kb + 4 * s + koff;
            v2f a = *(const v2f*)(Ap + k);           // ds_load_b64, conflict-free
            v2f b0, b1;
            b0.x = Wp0[(size_t)k * N_DIM];
            b0.y = Wp0[(size_t)(k + 1) * N_DIM];
            b1.x = Wp1[(size_t)k * N_DIM];
            b1.y = Wp1[(size_t)(k + 1) * N_DIM];
            // Two independent D = A(16x4)*B(4x16) + C chains in pure f32
            acc0 = __builtin_amdgcn_wmma_f32_16x16x4_f32(
                false, a, false, b0, (short)0, acc0, false, false);
            acc1 = __builtin_amdgcn_wmma_f32_16x16x4_f32(
                false, a, false, b1, (short)0, acc1, false, false);
        }
    }

    const float bv0 = bias[e * N_DIM + n0 + ncol];
    const float bv1 = bias[e * N_DIM + n0 + 16 + ncol];
#pragma unroll
    for (int i = 0; i < 8; ++i) {
        const int m   = i + ((lane >> 4) << 3);      // C/D: VGPR i -> M = i (+8 upper half)
        const int tok = s_tok[m];
        if (tok >= 0) {
            out[(size_t)tok * N_DIM + n0 + ncol]      = acc0[i] + bv0;
            out[(size_t)tok * N_DIM + n0 + 16 + ncol] = acc1[i] + bv1;
        }
    }
}

extern "C" void kernel_launch(void* const* d_in, const int* in_sizes, int n_in,
                              void* d_out, int out_size, void* d_ws, size_t ws_size,
                              hipStream_t stream) {
    const float* x      = (const float*)d_in[0];
    const int*   index  = (const int*)d_in[1];   // jax default x64-off => int32
    const float* weight = (const float*)d_in[2];
    const float* bias   = (const float*)d_in[3];
    float*       out    = (float*)d_out;

    int* wsi    = (int*)d_ws;
    int* off    = wsi;
    int* cursor = wsi + 16;
    int* counts = wsi + 32;
    int* perm   = wsi + 64;

    moe_init   <<<(PERM_CAP + 255) / 256, 256, 0, stream>>>(counts, perm);
    moe_count  <<<(N_TOKENS + 255) / 256, 256, 0, stream>>>(index, counts);
    moe_prefix <<<1, 32, 0, stream>>>(wsi);
    moe_scatter<<<(N_TOKENS + 255) / 256, 256, 0, stream>>>(index, cursor, perm);

    dim3 grid(MAX_TILES, N_DIM / 256, 1);
    moe_gemm<<<grid, 256, 0, stream>>>(x, weight, bias, off, perm, out);
}